// DoubleResNetUpDown_884763263428
// MI455X (gfx1250) — compile-verified
//
#include <hip/hip_runtime.h>
#include <stdint.h>

// Bandwidth-bound streaming kernel for 30x UpDownDoubleResNet blocks.
// Data path: global --(global_load_async_to_lds_b128, NT)--> LDS --> regs
//            --> 30-layer register chain --> LDS
//            --(global_store_async_from_lds_b128, NT)--> global
// Roofline: ~201 MB traffic vs ~3.1 GFLOP -> pure HBM-bandwidth-bound; f32 VALU
// is the right compute path (WMMA tile marshalling would only add overhead on
// a serial 30-layer dependence chain with 1x5 / 5x1 "matrices").

#define NLAYERS 30
#define BLOCK 256
#define RPT 4                       // rows per thread
#define TILE_ROWS (BLOCK * RPT)     // 1024 rows per block

__global__ __launch_bounds__(BLOCK)
void updown_tile_kernel(const float* __restrict__ x1, const float* __restrict__ x2,
                        const float* __restrict__ W1, const float* __restrict__ b1,
                        const float* __restrict__ W2, float* __restrict__ out, int nRows)
{
    __shared__ float sInX1[TILE_ROWS];        //  4 KB
    __shared__ float sInX2[TILE_ROWS * 5];    // 20 KB
    __shared__ float sOutX1[TILE_ROWS];       //  4 KB
    __shared__ float sOutX2[TILE_ROWS * 5];   // 20 KB
    __shared__ float sW1[NLAYERS * 5];
    __shared__ float sB1[NLAYERS];
    __shared__ float sW2[NLAYERS * 5];

    const int t = threadIdx.x;
    const long long tileBase = (long long)blockIdx.x * TILE_ROWS;

    // ---- Issue the bulk async global->LDS tile fetch FIRST (long latency),
    //      non-temporal: this data is touched exactly once device-wide. ----
    {
        uint32_t d1 = (uint32_t)(uintptr_t)(&sInX1[4 * t]);
        uint64_t g1 = (uint64_t)(uintptr_t)(x1 + tileBase + 4 * t);
        asm volatile("global_load_async_to_lds_b128 %0, %1, off th:TH_LOAD_NT"
                     :: "v"(d1), "v"(g1) : "memory");
#pragma unroll
        for (int j = 0; j < 5; ++j) {
            uint32_t d2 = (uint32_t)(uintptr_t)(&sInX2[4 * (t + BLOCK * j)]);
            uint64_t g2 = (uint64_t)(uintptr_t)(x2 + tileBase * 5 + 4 * (t + BLOCK * j));
            asm volatile("global_load_async_to_lds_b128 %0, %1, off th:TH_LOAD_NT"
                         :: "v"(d2), "v"(g2) : "memory");
        }
    }

    // ---- Overlap: stage per-layer params into LDS while the tile is in flight.
    //      Params are reused by every block -> leave them regular-temporal. ----
    for (int i = t; i < NLAYERS * 5; i += BLOCK) sW1[i] = W1[i];
    for (int i = t; i < NLAYERS;     i += BLOCK) sB1[i] = b1[i];
    for (int i = t; i < NLAYERS * 5; i += BLOCK) sW2[i] = W2[i];

    asm volatile("s_wait_asynccnt 0x0" ::: "memory");
    __syncthreads();

    // ---- Pull 4 rows into registers (stride-5 b32 LDS reads: conflict-free,
    //      gcd(5,64)=1 so 32 lanes hit 32 distinct banks). ----
    float a[RPT], v[RPT][5];
#pragma unroll
    for (int r = 0; r < RPT; ++r) {
        const int row = t + BLOCK * r;
        a[r] = sInX1[row];
#pragma unroll
        for (int k = 0; k < 5; ++k) v[r][k] = sInX2[row * 5 + k];
    }

    // ---- 30-layer chain entirely in registers; params broadcast from LDS ----
    for (int l = 0; l < NLAYERS; ++l) {
        float w1k[5], w2k[5];
#pragma unroll
        for (int k = 0; k < 5; ++k) { w1k[k] = sW1[l * 5 + k]; w2k[k] = sW2[l * 5 + k]; }
        const float bb = sB1[l];
#pragma unroll
        for (int r = 0; r < RPT; ++r) {
            float acc = a[r];
#pragma unroll
            for (int k = 0; k < 5; ++k)
                acc = fmaf(fmaxf(v[r][k], 0.0f), w1k[k], acc);
            acc += bb;
            a[r] = acc;
            const float ra = fmaxf(acc, 0.0f);
#pragma unroll
            for (int k = 0; k < 5; ++k)
                v[r][k] = fmaf(ra, w2k[k], v[r][k]);
        }
    }

    // ---- Results back to LDS ----
#pragma unroll
    for (int r = 0; r < RPT; ++r) {
        const int row = t + BLOCK * r;
        sOutX1[row] = a[r];
#pragma unroll
        for (int k = 0; k < 5; ++k) sOutX2[row * 5 + k] = v[r][k];
    }
    __syncthreads();  // LDS writes complete & visible before async engine reads them

    // ---- Async LDS -> global stores (128-bit fully coalesced, non-temporal) ----
    {
        float* outX1 = out;                 // [nRows]
        float* outX2 = out + nRows;         // [nRows*5]
        uint32_t s1 = (uint32_t)(uintptr_t)(&sOutX1[4 * t]);
        uint64_t g1 = (uint64_t)(uintptr_t)(outX1 + tileBase + 4 * t);
        asm volatile("global_store_async_from_lds_b128 %0, %1, off th:TH_STORE_NT"
                     :: "v"(g1), "v"(s1) : "memory");
#pragma unroll
        for (int j = 0; j < 5; ++j) {
            uint32_t s2 = (uint32_t)(uintptr_t)(&sOutX2[4 * (t + BLOCK * j)]);
            uint64_t g2 = (uint64_t)(uintptr_t)(outX2 + tileBase * 5 + 4 * (t + BLOCK * j));
            asm volatile("global_store_async_from_lds_b128 %0, %1, off th:TH_STORE_NT"
                         :: "v"(g2), "v"(s2) : "memory");
        }
    }
    // s_endpgm performs an implicit wait-idle covering outstanding ASYNCcnt.
}

// Scalar tail for rows not covered by full tiles (not hit for N = 4,194,304).
__global__ void updown_tail_kernel(const float* __restrict__ x1, const float* __restrict__ x2,
                                   const float* __restrict__ W1, const float* __restrict__ b1,
                                   const float* __restrict__ W2, float* __restrict__ out,
                                   int startRow, int nRows)
{
    int row = startRow + blockIdx.x * blockDim.x + threadIdx.x;
    if (row >= nRows) return;
    float a = x1[row];
    float v0 = x2[row * 5 + 0], v1 = x2[row * 5 + 1], v2 = x2[row * 5 + 2],
          v3 = x2[row * 5 + 3], v4 = x2[row * 5 + 4];
    for (int l = 0; l < NLAYERS; ++l) {
        float acc = a;
        acc = fmaf(fmaxf(v0, 0.f), W1[l * 5 + 0], acc);
        acc = fmaf(fmaxf(v1, 0.f), W1[l * 5 + 1], acc);
        acc = fmaf(fmaxf(v2, 0.f), W1[l * 5 + 2], acc);
        acc = fmaf(fmaxf(v3, 0.f), W1[l * 5 + 3], acc);
        acc = fmaf(fmaxf(v4, 0.f), W1[l * 5 + 4], acc);
        acc += b1[l];
        a = acc;
        float ra = fmaxf(acc, 0.f);
        v0 = fmaf(ra, W2[l * 5 + 0], v0);
        v1 = fmaf(ra, W2[l * 5 + 1], v1);
        v2 = fmaf(ra, W2[l * 5 + 2], v2);
        v3 = fmaf(ra, W2[l * 5 + 3], v3);
        v4 = fmaf(ra, W2[l * 5 + 4], v4);
    }
    out[row] = a;
    float* o2 = out + nRows;
    o2[row * 5 + 0] = v0; o2[row * 5 + 1] = v1; o2[row * 5 + 2] = v2;
    o2[row * 5 + 3] = v3; o2[row * 5 + 4] = v4;
}

extern "C" void kernel_launch(void* const* d_in, const int* in_sizes, int n_in,
                              void* d_out, int out_size, void* d_ws, size_t ws_size,
                              hipStream_t stream)
{
    const float* x1 = (const float*)d_in[0];
    const float* x2 = (const float*)d_in[1];
    const float* W1 = (const float*)d_in[2];
    const float* b1 = (const float*)d_in[3];
    const float* W2 = (const float*)d_in[4];
    float* out = (float*)d_out;
    const int nRows = in_sizes[0];

    const int nTiles = nRows / TILE_ROWS;
    if (nTiles > 0) {
        updown_tile_kernel<<<nTiles, BLOCK, 0, stream>>>(x1, x2, W1, b1, W2, out, nRows);
    }
    const int done = nTiles * TILE_ROWS;
    const int rem = nRows - done;
    if (rem > 0) {
        int blocks = (rem + BLOCK - 1) / BLOCK;
        updown_tail_kernel<<<blocks, BLOCK, 0, stream>>>(x1, x2, W1, b1, W2, out, done, nRows);
    }
}